// SparseConvLayer_28930899706435
// MI455X (gfx1250) — compile-verified
//
#include <hip/hip_runtime.h>
#include <hip/hip_bf16.h>

// ---------------------------------------------------------------------------
// Sparse conv = scatter-add densify (NHWC f32 in d_ws) + implicit-GEMM 3x3
// conv done with V_WMMA_F32_16X16X4_F32 (faithful f32 path on gfx1250).
// ---------------------------------------------------------------------------

typedef __attribute__((ext_vector_type(2))) float v2f;
typedef __attribute__((ext_vector_type(4))) float v4f;
typedef __attribute__((ext_vector_type(8))) float v8f;

#define HH 512
#define WW 512
#define CIN 64
#define COUT 128
#define BATCH 4

// ---- zero the dense NHWC scratch grid (grid-stride float4 stores) ----------
__global__ __launch_bounds__(256) void zero_dense_kernel(v4f* __restrict__ p, long n4) {
    long stride = (long)gridDim.x * blockDim.x;
    v4f z = {0.f, 0.f, 0.f, 0.f};
    for (long i = (long)blockIdx.x * blockDim.x + threadIdx.x; i < n4; i += stride)
        p[i] = z;
}

// ---- reorganize OIHW weights into WMMA B-fragment order --------------------
// Flat layout: idx = ((tap*16 + cin_blk)*8 + co_blk)*64 + lane*2 + v
//   lane l: co = co_blk*16 + (l&15), cin = cin_blk*4 + (l>>4)*2 + v
// so each wave's B-fragment load is one contiguous 256B b64 load.
__global__ __launch_bounds__(256) void wreorg_kernel(const float* __restrict__ w,
                                                     float* __restrict__ wR) {
    int idx = blockIdx.x * blockDim.x + threadIdx.x;
    if (idx >= 9 * 16 * 8 * 64) return;
    int v    = idx & 1;
    int l    = (idx >> 1) & 31;
    int blk  = idx >> 6;
    int cob  = blk & 7;
    int cinb = (blk >> 3) & 15;
    int tap  = blk >> 7;
    int co   = cob * 16 + (l & 15);
    int cin  = cinb * 4 + ((l >> 4) << 1) + v;
    int ky   = tap / 3, kx = tap % 3;
    wR[idx] = w[((co * CIN + cin) * 3 + ky) * 3 + kx];
}

// ---- scatter-add point features onto dense NHWC grid -----------------------
__global__ __launch_bounds__(256) void scatter_kernel(const float* __restrict__ feat,
                                                      const int* __restrict__ coors,
                                                      float* __restrict__ dense, int npts) {
    long t = (long)blockIdx.x * blockDim.x + threadIdx.x;
    int pt = (int)(t >> 6);
    int c  = (int)(t & 63);
    if (pt >= npts) return;
    int b  = coors[pt * 3 + 0];
    int yy = coors[pt * 3 + 1];
    int xx = coors[pt * 3 + 2];
    long didx = ((((long)b * HH + yy) * WW + xx) << 6) + c;  // NHWC
    atomicAdd(&dense[didx], feat[(long)pt * CIN + c]);
}

// ---- implicit-GEMM 3x3 conv via V_WMMA_F32_16X16X4_F32 ---------------------
// One wave: 32 output pixels (two 16-px subtiles) x all 128 co (8 co-blocks)
// for one (b, y). 9 taps x 16 cin-blocks K-loop, 16 WMMAs per K-step.
__global__ __launch_bounds__(256) void sconv_wmma_kernel(const float* __restrict__ dense,
                                                         const float* __restrict__ wR,
                                                         float* __restrict__ out) {
    const int lane = threadIdx.x & 31;
    const int wid  = blockIdx.x * 8 + (threadIdx.x >> 5); // 8 waves per block
    const int xt   = wid & 15;          // 16 x-tiles of 32 px
    const int y    = (wid >> 4) & 511;
    const int b    = wid >> 13;
    const int x0   = xt << 5;

    v8f acc[8][2];
#pragma unroll
    for (int cb = 0; cb < 8; ++cb)
#pragma unroll
        for (int s = 0; s < 2; ++s)
#pragma unroll
            for (int e = 0; e < 8; ++e)
                acc[cb][s][e] = 0.0f;

    const int lx = lane & 15;           // A: M = pixel within subtile
    const int kc = (lane >> 4) << 1;    // A: K components 0,1 (lanes<16) / 2,3

    for (int ky = 0; ky < 3; ++ky) {
        const int yy = y + ky - 1;
        const bool rowok = (unsigned)yy < (unsigned)HH;
        const float* rowp = dense + ((long)(b * HH + (rowok ? yy : 0)) << 15); // *W*C
        for (int kx = 0; kx < 3; ++kx) {
            const int tap = ky * 3 + kx;
            const int xx0 = x0 + lx + kx - 1;
            const int xx1 = xx0 + 16;
            const bool ok0 = rowok && ((unsigned)xx0 < (unsigned)WW);
            const bool ok1 = rowok && ((unsigned)xx1 < (unsigned)WW);
            const float* p0 = rowp + ((ok0 ? xx0 : 0) << 6) + kc;
            const float* p1 = rowp + ((ok1 ? xx1 : 0) << 6) + kc;
            const float* wp = wR + ((long)tap << 13) + (lane << 1); // tap*16*8*64
            for (int cinb = 0; cinb < 16; ++cinb) {
                v2f a0 = {0.f, 0.f};
                v2f a1 = {0.f, 0.f};
                if (ok0) a0 = *(const v2f*)(p0 + (cinb << 2));
                if (ok1) a1 = *(const v2f*)(p1 + (cinb << 2));
#pragma unroll
                for (int cb = 0; cb < 8; ++cb) {
                    v2f wf = *(const v2f*)(wp + (((cinb << 3) + cb) << 6));
                    acc[cb][0] = __builtin_amdgcn_wmma_f32_16x16x4_f32(
                        false, a0, false, wf, (short)0, acc[cb][0], false, false);
                    acc[cb][1] = __builtin_amdgcn_wmma_f32_16x16x4_f32(
                        false, a1, false, wf, (short)0, acc[cb][1], false, false);
                }
            }
        }
    }

    // Store: D layout -> lane holds 8 x-contiguous floats for one co.
    const int co_l = lane & 15;
    const int pxo  = (lane >> 4) << 3;  // 0 or 8
#pragma unroll
    for (int cb = 0; cb < 8; ++cb) {
        const int co = (cb << 4) + co_l;
        float* obase = out + ((((long)(b * COUT + co)) << 9) + y) * WW + x0; // NCHW
#pragma unroll
        for (int s = 0; s < 2; ++s) {
            v4f lo = {acc[cb][s][0], acc[cb][s][1], acc[cb][s][2], acc[cb][s][3]};
            v4f hi = {acc[cb][s][4], acc[cb][s][5], acc[cb][s][6], acc[cb][s][7]};
            float* p = obase + (s << 4) + pxo;
            __builtin_nontemporal_store(lo, (v4f*)p);        // stream 512MB out,
            __builtin_nontemporal_store(hi, (v4f*)(p + 4));  // keep dense in L2
        }
    }
}

extern "C" void kernel_launch(void* const* d_in, const int* in_sizes, int n_in,
                              void* d_out, int out_size, void* d_ws, size_t ws_size,
                              hipStream_t stream) {
    const float* feat  = (const float*)d_in[0];
    const int*   coors = (const int*)d_in[1];
    // d_in[2] = batch_size scalar (==4, baked into constants)
    const float* w     = (const float*)d_in[3];
    const int npts = in_sizes[0] / CIN;

    float* dense = (float*)d_ws;                               // 4*512*512*64 f32 = 256MB
    float* wR    = dense + (long)BATCH * HH * WW * CIN;        // +288KB reorg weights

    long n4 = ((long)BATCH * HH * WW * CIN) / 4;
    zero_dense_kernel<<<4096, 256, 0, stream>>>((v4f*)dense, n4);

    wreorg_kernel<<<(9 * 16 * 8 * 64 + 255) / 256, 256, 0, stream>>>(w, wR);

    long nthreads = (long)npts * CIN;
    scatter_kernel<<<(int)((nthreads + 255) / 256), 256, 0, stream>>>(feat, coors, dense, npts);

    // 4 batch * 512 rows * 16 x-tiles = 32768 wave-tiles / 8 waves per block
    sconv_wmma_kernel<<<4096, 256, 0, stream>>>(dense, wR, (float*)d_out);
}